// SubspaceLinopFactory_75127567941921
// MI455X (gfx1250) — compile-verified
//
#include <hip/hip_runtime.h>
#include <math.h>

typedef __attribute__((ext_vector_type(2))) float v2f;
typedef __attribute__((ext_vector_type(8))) float v8f;

#define A_  4
#define T_  64
#define C_  8
#define K_  1024
#define R_  8
#define HH  64
#define WW  64
#define HW  4096   // HH*WW
#define AC  32     // A_*C_
#define KT  32     // k-columns per workgroup tile (2 WMMA N-tiles)

// D = A(16x4,f32) * B(4x16,f32) + C(16x16,f32), wave32 WMMA
__device__ __forceinline__ v8f wmma_f32(v2f a, v2f b, v8f c) {
    return __builtin_amdgcn_wmma_f32_16x16x4_f32(
        /*neg_a=*/false, a, /*neg_b=*/false, b,
        /*c_mod=*/(short)0, c, /*reuse_a=*/false, /*reuse_b=*/false);
}

// ---------------------------------------------------------------------------
// Kernel 1: coil images as separate Re/Im planes, [ac][hw] row-major so a
// single global_load_b64 yields a complete WMMA A-fragment (K=hw, K=hw+1).
// ---------------------------------------------------------------------------
__global__ void build_img(const float* __restrict__ xr, const float* __restrict__ xi,
                          const float* __restrict__ mr, const float* __restrict__ mi,
                          float* __restrict__ imgRe, float* __restrict__ imgIm) {
    int tid = blockIdx.x * blockDim.x + threadIdx.x;
    if (tid >= AC * HW) return;
    int hw = tid & (HW - 1);
    int ac = tid >> 12;
    int a = ac >> 3;           // ac = a*C_ + c
    int c = ac & (C_ - 1);
    float x_r = xr[a * HW + hw], x_i = xi[a * HW + hw];
    float m_r = mr[c * HW + hw], m_i = mi[c * HW + hw];
    imgRe[ac * HW + hw] = x_r * m_r - x_i * m_i;
    imgIm[ac * HW + hw] = x_r * m_i + x_i * m_r;
}

// ---------------------------------------------------------------------------
// Kernel 2: exact NDFT as complex GEMM  F[r,ac,k] = img(32x4096) x E2^T(4096x32)
// E2[hw,k] = Eh[k,h]*Ew[k,w] generated on the fly from LDS tiles.
// One workgroup per (r, 32-wide k tile); 4 waves split the hw inner dimension;
// each wave carries 2 M-tiles x 2 N-tiles so every A-fragment feeds 4 WMMAs.
// Ew tile is stored PAIR-SPLIT: float4 = (re0, re1, im0, im1) per w-pair so the
// b128 LDS load lands directly in v_pk_* source-pair registers (no shuffles).
// ---------------------------------------------------------------------------
__global__ __launch_bounds__(128)
void ndft_wmma_kernel(const float* __restrict__ trj,
                      const float* __restrict__ imgRe,
                      const float* __restrict__ imgIm,
                      float2* __restrict__ Fws) {
    // 32 KB union: [phase A/B] Eh (16KB) + Ew (16KB)  /  [phase C] partials (32KB)
    __shared__ float4 smemBuf[2048];
    float2* EhT  = (float2*)smemBuf;       // Eh[klocal][h],  klocal 0..31
    float4* EwT4 = smemBuf + 1024;         // Ew[klocal][wpair] = (re0,re1,im0,im1)
    float2* partB = (float2*)smemBuf;      // part[wave][m][n] after barrier

    const int bid = blockIdx.x;
    const int r   = bid >> 5;              // K_/KT = 32 tiles per r
    const int k0  = (bid & 31) * KT;
    const int tid = threadIdx.x;

    // ---- Phase A: build Eh/Ew tiles (centered grid, exp(-i*trj*g)) --------
    // entries 0..2047: Eh[n][g] (float2);  2048..3071: Ew[n][wpair] (float4)
    for (int e = tid; e < KT * HH + KT * (WW / 2); e += 128) {
        if (e < KT * HH) {
            int n = e >> 6;
            int g = e & 63;
            float ang = -trj[(r * 2 + 0) * K_ + k0 + n] * (float)(g - 32);
            float s, c;
            __sincosf(ang, &s, &c);
            EhT[n * 64 + g] = make_float2(c, s);
        } else {
            int p  = e - KT * HH;
            int n  = p >> 5;
            int wp = p & 31;
            float tw   = trj[(r * 2 + 1) * K_ + k0 + n];
            float ang0 = -tw * (float)(2 * wp - 32);
            float ang1 = ang0 - tw;
            float s0, c0, s1, c1;
            __sincosf(ang0, &s0, &c0);
            __sincosf(ang1, &s1, &c1);
            EwT4[n * 32 + wp] = make_float4(c0, c1, s0, s1);   // (re0,re1,im0,im1)
        }
    }
    __syncthreads();

    // ---- Phase B: WMMA main loop ------------------------------------------
    const int lane = tid & 31;
    const int wave = tid >> 5;
    const int kk   = (lane >> 4) << 1;     // 0 (lanes 0-15) or 2 (lanes 16-31)
    const int mrow = lane & 15;            // A-frag row
    const int n    = lane & 15;            // B/C column within each N-tile

    const float* ar0p = imgRe + mrow * HW;          // rows 0..15
    const float* ai0p = imgIm + mrow * HW;
    const float* ar1p = imgRe + (mrow + 16) * HW;   // rows 16..31
    const float* ai1p = imgIm + (mrow + 16) * HW;

    v8f zz = {0,0,0,0,0,0,0,0};
    v8f c0reA = zz, c0imA = zz, c1reA = zz, c1imA = zz;   // N-tile A (k0..k0+15)
    v8f c0reB = zz, c0imB = zz, c1reB = zz, c1imB = zz;   // N-tile B (k0+16..k0+31)

    // each wave owns 16 h-rows; eh is invariant across the 64-wide w sweep
    for (int hl = 0; hl < 16; ++hl) {
        const int h = (wave << 4) + hl;
        const float2 ehA = EhT[n * 64 + h];
        const float2 ehB = EhT[(n + 16) * 64 + h];
        const int hwRow = h << 6;

        // speculative prefetch of the next h-row of the A stream
        __builtin_prefetch(ar0p + hwRow + 64, 0, 0);
        __builtin_prefetch(ai0p + hwRow + 64, 0, 0);

        #pragma unroll 4
        for (int sl = 0; sl < 16; ++sl) {
            const int w0  = (sl << 2) + kk;          // even
            const int hw0 = hwRow + w0;
            const int wp  = w0 >> 1;

            // A fragments: direct b64 loads, reused by all 16 WMMAs below
            v2f ar0 = *(const v2f*)(ar0p + hw0);
            v2f ai0 = *(const v2f*)(ai0p + hw0);
            v2f ar1 = *(const v2f*)(ar1p + hw0);
            v2f ai1 = *(const v2f*)(ai1p + hw0);

            // B fragments, N-tile A: (ew.x,ew.y)=re pair, (ew.z,ew.w)=im pair
            float4 ewA = EwT4[(n << 5) + wp];
            v2f brA, biA, bnA;
            brA.x = ehA.x * ewA.x - ehA.y * ewA.z;
            brA.y = ehA.x * ewA.y - ehA.y * ewA.w;
            biA.x = ehA.x * ewA.z + ehA.y * ewA.x;
            biA.y = ehA.x * ewA.w + ehA.y * ewA.y;
            bnA.x = -biA.x;  bnA.y = -biA.y;

            c0reA = wmma_f32(ar0, brA, c0reA);
            c0reA = wmma_f32(ai0, bnA, c0reA);
            c0imA = wmma_f32(ar0, biA, c0imA);
            c0imA = wmma_f32(ai0, brA, c0imA);
            c1reA = wmma_f32(ar1, brA, c1reA);
            c1reA = wmma_f32(ai1, bnA, c1reA);
            c1imA = wmma_f32(ar1, biA, c1imA);
            c1imA = wmma_f32(ai1, brA, c1imA);

            // B fragments, N-tile B (columns k0+16 .. k0+31)
            float4 ewB = EwT4[((n + 16) << 5) + wp];
            v2f brB, biB, bnB;
            brB.x = ehB.x * ewB.x - ehB.y * ewB.z;
            brB.y = ehB.x * ewB.y - ehB.y * ewB.w;
            biB.x = ehB.x * ewB.z + ehB.y * ewB.x;
            biB.y = ehB.x * ewB.w + ehB.y * ewB.y;
            bnB.x = -biB.x;  bnB.y = -biB.y;

            c0reB = wmma_f32(ar0, brB, c0reB);
            c0reB = wmma_f32(ai0, bnB, c0reB);
            c0imB = wmma_f32(ar0, biB, c0imB);
            c0imB = wmma_f32(ai0, brB, c0imB);
            c1reB = wmma_f32(ar1, brB, c1reB);
            c1reB = wmma_f32(ai1, bnB, c1reB);
            c1imB = wmma_f32(ar1, biB, c1imB);
            c1imB = wmma_f32(ai1, brB, c1imB);
        }
    }

    // E-tiles are dead from here; reuse the 32KB LDS for partial tiles.
    __syncthreads();

    // ---- Phase C: cross-wave reduction via LDS ----------------------------
    // 16x16 f32 C/D layout: VGPR j holds M = j + 8*(lane>=16), N = lane&15
    {
        const int mOff = (lane >> 4) * 8;
        #pragma unroll
        for (int j = 0; j < 8; ++j) {
            int m = mOff + j;
            float2* pw = partB + (wave * AC) * KT;
            pw[m * KT + n]              = make_float2(c0reA[j], c0imA[j]);
            pw[m * KT + n + 16]         = make_float2(c0reB[j], c0imB[j]);
            pw[(m + 16) * KT + n]       = make_float2(c1reA[j], c1imA[j]);
            pw[(m + 16) * KT + n + 16]  = make_float2(c1reB[j], c1imB[j]);
        }
    }
    __syncthreads();
    for (int e = tid; e < AC * KT; e += 128) {
        int m = e >> 5, nn = e & 31;
        float2 s0 = partB[(0 * AC + m) * KT + nn];
        float2 s1 = partB[(1 * AC + m) * KT + nn];
        float2 s2 = partB[(2 * AC + m) * KT + nn];
        float2 s3 = partB[(3 * AC + m) * KT + nn];
        Fws[(r * AC + m) * K_ + k0 + nn] =
            make_float2(s0.x + s1.x + s2.x + s3.x,
                        s0.y + s1.y + s2.y + s3.y);
    }
}

// ---------------------------------------------------------------------------
// Kernel 3: gather trajectories + phi/dcf combine -> y[i,t,c,k] complex64
// ---------------------------------------------------------------------------
__global__ void combine_kernel(const float* __restrict__ phi,
                               const float* __restrict__ sqrt_dcf,
                               const int* __restrict__ sub,
                               const float2* __restrict__ Fws,
                               float2* __restrict__ out, int I) {
    int tid = blockIdx.x * blockDim.x + threadIdx.x;
    int total = I * T_ * C_ * K_;
    if (tid >= total) return;
    int k = tid & (K_ - 1);
    int c = (tid >> 10) & (C_ - 1);
    int t = (tid >> 13) & (T_ - 1);
    int i = tid >> 19;
    int r = sub[i * T_ + t];
    float accr = 0.f, acci = 0.f;
    #pragma unroll
    for (int a = 0; a < A_; ++a) {
        float p = phi[a * T_ + t];
        float2 f = Fws[(r * AC + a * C_ + c) * K_ + k];
        accr += p * f.x;
        acci += p * f.y;
    }
    float sc = 0.03125f * sqrt_dcf[r * K_ + k];  // OVERSAMP/sqrt(H*W) = 2/64
    out[tid] = make_float2(sc * accr, sc * acci);
}

// ---------------------------------------------------------------------------
extern "C" void kernel_launch(void* const* d_in, const int* in_sizes, int n_in,
                              void* d_out, int out_size, void* d_ws, size_t ws_size,
                              hipStream_t stream) {
    (void)n_in; (void)out_size; (void)ws_size;
    const float* x_re     = (const float*)d_in[0];
    const float* x_im     = (const float*)d_in[1];
    const float* mps_re   = (const float*)d_in[2];
    const float* mps_im   = (const float*)d_in[3];
    const float* phi      = (const float*)d_in[4];
    const float* sqrt_dcf = (const float*)d_in[5];
    const float* trj      = (const float*)d_in[6];
    const int*   sub      = (const int*)d_in[7];
    int I = in_sizes[7] / T_;

    float*  imgRe = (float*)d_ws;                                   // 512 KB
    float*  imgIm = imgRe + (size_t)AC * HW;                        // 512 KB
    float2* Fws   = (float2*)(imgIm + (size_t)AC * HW);             // 2 MB

    build_img<<<(AC * HW + 255) / 256, 256, 0, stream>>>(x_re, x_im, mps_re, mps_im,
                                                         imgRe, imgIm);
    ndft_wmma_kernel<<<R_ * (K_ / KT), 128, 0, stream>>>(trj, imgRe, imgIm, Fws);
    int total = I * T_ * C_ * K_;
    combine_kernel<<<(total + 255) / 256, 256, 0, stream>>>(phi, sqrt_dcf, sub, Fws,
                                                            (float2*)d_out, I);
}